// LSH_6339371728928
// MI455X (gfx1250) — compile-verified
//
#include <hip/hip_runtime.h>
#include <stdint.h>

// ret[o] = sum_{s,a} x[s] * w[o,s,a]
// x: [4096] f32, w: [512, 4096, 128] f32 (1 GiB), out: [512] f32.
// Memory-bound streaming reduction: floor = 1GiB / 23.3TB/s ~= 46 us.
// gfx1250 paths exercised:
//   - global_load_async_to_lds_b128 (ASYNCcnt) 4-deep ring, 2 stages in
//     flight, th:TH_LOAD_NT so the one-shot 1GiB stream does not evict L2
//   - s_wait_asynccnt split-counter throttling
//   - ds_load_b128 consume + VALU fma (main loop stays VALU: f32 WMMA in the
//     inner loop would need ~186 TF f32 matrix rate to track 23.3 TB/s)
//   - v_wmma_f32_16x16x4_f32 epilogue: ones-matrix column-sum collapses the
//     32 per-lane partials in one matrix op (free: once per 32 KiB streamed)

#define S_DIM   4096
#define A_DIM   128
#define O_DIM   512
#define SPLIT   8                         // s-chunks per output -> 4096 blocks
#define WAVES   8                         // 256 threads / wave32
#define UNROLL  4                         // async b128 instrs per stage per wave
#define NBUF    4                         // LDS ring depth (2-stage lookahead)

#define S_PER_BLOCK (S_DIM / SPLIT)                 // 512
#define S_PER_WAVE  (S_PER_BLOCK / WAVES)           // 64
#define FLOATS_PER_WAVE (S_PER_WAVE * A_DIM)        // 8192 (32 KiB / wave)
#define STAGE_FLOATS (32 * 4 * UNROLL)              // 512 floats = UNROLL s-rows
#define NSTAGES (FLOATS_PER_WAVE / STAGE_FLOATS)    // 16

#define WAIT_ASYNC(N) asm volatile("s_wait_asynccnt " #N ::: "memory")

typedef __attribute__((ext_vector_type(2))) float v2f;
typedef __attribute__((ext_vector_type(8))) float v8f;

__global__ __launch_bounds__(256) void lsh_reduce_kernel(
        const float* __restrict__ x,
        const float* __restrict__ w,
        float* __restrict__ out)
{
    __shared__ float  xs[S_PER_BLOCK];                    // 2 KiB x chunk
    __shared__ float4 stage[WAVES][NBUF][32 * UNROLL];    // 64 KiB ring buffers

    const int o   = blockIdx.x;          // output index
    const int c   = blockIdx.y;          // s-chunk index
    const int tid = threadIdx.x;
    const int wv  = tid >> 5;            // wave id (wave32)
    const int ln  = tid & 31;            // lane id

    // Stage the x sub-range this block needs into LDS (broadcast table).
    for (int i = tid; i < S_PER_BLOCK; i += 256)
        xs[i] = x[c * S_PER_BLOCK + i];
    __syncthreads();

    // Each wave streams a contiguous 32 KiB slab of w, perfectly coalesced:
    // one async B128 instruction = 32 lanes x 16B = one full attribute row,
    // so the x[s] scale is wave-uniform per instruction.
    const float* wave_ptr = w + (size_t)o * (S_DIM * A_DIM)
                              + (size_t)c * (S_PER_BLOCK * A_DIM)
                              + (size_t)wv * FLOATS_PER_WAVE;

    auto issue_stage = [&](int t, int buf) {
        #pragma unroll
        for (int u = 0; u < UNROLL; ++u) {
            // LDS byte offset of this lane's slot (flat addr low 32 bits).
            uint32_t lds_off = (uint32_t)(uintptr_t)&stage[wv][buf][u * 32 + ln];
            uint64_t gaddr   = (uint64_t)(uintptr_t)(wave_ptr + (size_t)t * STAGE_FLOATS
                                                     + u * A_DIM + ln * 4);
            asm volatile("global_load_async_to_lds_b128 %0, %1, off th:TH_LOAD_NT"
                         :: "v"(lds_off), "v"(gaddr)
                         : "memory");
        }
    };

    float acc = 0.0f;

    // Prologue: two stages in flight.
    issue_stage(0, 0);
    issue_stage(1, 1);

    for (int t = 0; t < NSTAGES; ++t) {
        const int cur = t & (NBUF - 1);
        if (t + 2 < NSTAGES) {
            issue_stage(t + 2, (t + 2) & (NBUF - 1));
            WAIT_ASYNC(8);   // stage t done; t+1, t+2 (8 instrs) still in flight
        } else if (t + 1 < NSTAGES) {
            WAIT_ASYNC(4);   // stage t done; t+1 in flight
        } else {
            WAIT_ASYNC(0);   // drain
        }
        #pragma unroll
        for (int u = 0; u < UNROLL; ++u) {
            float4 v = stage[wv][cur][u * 32 + ln];            // ds_load_b128
            float  sc = xs[wv * S_PER_WAVE + t * UNROLL + u];  // LDS broadcast
            acc = fmaf(sc, (v.x + v.y) + (v.z + v.w), acc);
        }
    }

    // ---- Epilogue reduction: 32 lane partials -> 1 scalar per wave. ----
    float r;
#if __has_builtin(__builtin_amdgcn_wmma_f32_16x16x4_f32)
    // D = A(ones,16x4) x B(4x16) + 0  =>  D[m][n] = sum_k B[k][n], all rows
    // equal. Each lane contributes its partial once (second B VGPR = 0), so
    // the 64 B-slots hold each acc exactly once -> every D row is the sum of
    // a disjoint pair of lanes' partials per column. Summing D's first VGPR
    // over all 32 lanes counts every column twice -> scale by 0.5.
    {
        v2f a_ones = {1.0f, 1.0f};
        v2f b_acc  = {acc, 0.0f};
        v8f c_zero = {};
        v8f d = __builtin_amdgcn_wmma_f32_16x16x4_f32(
            /*neg_a=*/false, a_ones, /*neg_b=*/false, b_acc,
            /*c_mod=*/(short)0, c_zero, /*reuse_a=*/false, /*reuse_b=*/false);
        r = d[0];
        #pragma unroll
        for (int off = 16; off > 0; off >>= 1)
            r += __shfl_xor(r, off, 32);
        r *= 0.5f;
    }
#else
    r = acc;
    #pragma unroll
    for (int off = 16; off > 0; off >>= 1)
        r += __shfl_xor(r, off, 32);
#endif

    if (ln == 0)
        atomicAdd(&out[o], r);   // 8 atomics/block, 512 target addresses
}

__global__ void zero_out_kernel(float* __restrict__ out, int n) {
    int i = blockIdx.x * blockDim.x + threadIdx.x;
    if (i < n) out[i] = 0.0f;
}

extern "C" void kernel_launch(void* const* d_in, const int* in_sizes, int n_in,
                              void* d_out, int out_size, void* d_ws, size_t ws_size,
                              hipStream_t stream) {
    const float* x = (const float*)d_in[0];   // 4096 floats
    const float* w = (const float*)d_in[1];   // 512*4096*128 floats
    float* out = (float*)d_out;               // 512 floats

    zero_out_kernel<<<(out_size + 255) / 256, 256, 0, stream>>>(out, out_size);

    dim3 grid(O_DIM, SPLIT);
    lsh_reduce_kernel<<<grid, 256, 0, stream>>>(x, w, out);
}